// NodeProcessor_31877247271255
// MI455X (gfx1250) — compile-verified
//
#include <hip/hip_runtime.h>

#define N_NODES 100000
#define N_EDGES 1200000
#define D_NODE 64
#define D_EDGE 64
#define HIDDEN 128
#define D_IN   128
#define LN_EPS 1e-5f
#define WAVES  4

typedef __attribute__((ext_vector_type(2))) float v2f;
typedef __attribute__((ext_vector_type(8))) float v8f;

// ---------------------------------------------------------------- zero agg
__global__ void zero_kernel(float4* __restrict__ p, int n4) {
    int i = blockIdx.x * blockDim.x + threadIdx.x;
    if (i < n4) p[i] = make_float4(0.f, 0.f, 0.f, 0.f);
}

// ---------------------------------------------------------------- scatter-sum
// One wave per edge; lane owns 2 of the 64 features. agg fits in L2, so the
// fp32 HW atomics (global_atomic_add_f32) resolve at L2 bandwidth.
__global__ __launch_bounds__(256) void scatter_kernel(
    const float* __restrict__ edge_attr,
    const long long* __restrict__ col,     // edge_index row 1
    float* __restrict__ agg) {
    int e    = (blockIdx.x * blockDim.x + threadIdx.x) >> 5;
    int lane = threadIdx.x & 31;
    if (e >= N_EDGES) return;
    long long c = col[e];
    const float2 v = *(const float2*)(edge_attr + (size_t)e * D_EDGE + lane * 2);
    float* dst = agg + (size_t)c * D_EDGE + lane * 2;
    unsafeAtomicAdd(dst + 0, v.x);
    unsafeAtomicAdd(dst + 1, v.y);
}

// ---------------------------------------------------------------- WMMA layer
// One wave computes tout[16 x NT*16] = act(tin[16x128] @ w[128 x NT*16] + bias)
// using V_WMMA_F32_16X16X4_F32.
//  A layout (16x4 fp32): lane m=lane&15; VGPR0/1 hold K = kb, kb+1 where
//  kb = (lane<16 ? 0 : 2)  -> ds_load_b64 of tin[m][k0+kb .. k0+kb+1].
//  B layout (4x16 fp32): n = lane&15; VGPR0/1 hold K = kb, kb+1 at column n.
//  C/D layout (16x16 fp32): VGPR r holds row (r + (lane<16?0:8)), col lane&15.
template<int NT, bool RELU>
__device__ __forceinline__ void layer128(const float* __restrict__ tin,
                                         const float* __restrict__ w,
                                         const float* __restrict__ bias,
                                         float* __restrict__ tout, int lane) {
    const int m   = lane & 15;
    const int kb  = (lane >> 4) << 1;   // 0 or 2
    const int mhi = (lane >> 4) << 3;   // 0 or 8
    const int NOUT = NT * 16;

    v2f afr[32];
#pragma unroll
    for (int k = 0; k < 32; ++k)
        afr[k] = *(const v2f*)(tin + m * 128 + k * 4 + kb);

#pragma unroll
    for (int nt = 0; nt < NT; ++nt) {
        float bv = bias[nt * 16 + m];
        v8f c = {bv, bv, bv, bv, bv, bv, bv, bv};
#pragma unroll
        for (int k = 0; k < 32; ++k) {
            const float* wp = w + (k * 4 + kb) * NOUT + nt * 16 + m;
            v2f b;
            b.x = wp[0];
            b.y = wp[NOUT];
            c = __builtin_amdgcn_wmma_f32_16x16x4_f32(
                    false, afr[k], false, b, (short)0, c, false, false);
        }
#pragma unroll
        for (int r = 0; r < 8; ++r) {
            float v = c[r];
            if (RELU) v = fmaxf(v, 0.f);
            tout[(r + mhi) * NOUT + nt * 16 + m] = v;
        }
    }
}

// ---------------------------------------------------------------- fused MLP
__global__ __launch_bounds__(32 * WAVES) void mlp_kernel(
    const float* __restrict__ x,  const float* __restrict__ agg,
    const float* __restrict__ W0, const float* __restrict__ b0,
    const float* __restrict__ W1, const float* __restrict__ b1,
    const float* __restrict__ W2, const float* __restrict__ b2,
    const float* __restrict__ lng, const float* __restrict__ lnb,
    float* __restrict__ out) {
    // All weights resident in LDS (~226 KB total < 320 KB/WGP).
    __shared__ float sW0[D_IN * HIDDEN];
    __shared__ float sW1[HIDDEN * HIDDEN];
    __shared__ float sW2[HIDDEN * D_NODE];
    __shared__ float sB0[HIDDEN];
    __shared__ float sB1[HIDDEN];
    __shared__ float sB2[D_NODE];
    __shared__ float sG[D_NODE];
    __shared__ float sBt[D_NODE];
    __shared__ float sTile[WAVES * 2 * 16 * 128];   // per-wave ping/pong tiles

    const int tid = threadIdx.x;
    for (int i = tid; i < D_IN * HIDDEN / 4; i += blockDim.x)
        ((float4*)sW0)[i] = ((const float4*)W0)[i];
    for (int i = tid; i < HIDDEN * HIDDEN / 4; i += blockDim.x)
        ((float4*)sW1)[i] = ((const float4*)W1)[i];
    for (int i = tid; i < HIDDEN * D_NODE / 4; i += blockDim.x)
        ((float4*)sW2)[i] = ((const float4*)W2)[i];
    if (tid < HIDDEN) { sB0[tid] = b0[tid]; sB1[tid] = b1[tid]; }
    if (tid < D_NODE) { sB2[tid] = b2[tid]; sG[tid] = lng[tid]; sBt[tid] = lnb[tid]; }
    __syncthreads();

    const int wid  = tid >> 5;
    const int lane = tid & 31;
    float* tA = sTile + wid * (2 * 16 * 128);
    float* tB = tA + 16 * 128;

    const int nTiles = N_NODES / 16;   // 6250, exact
    for (int t = blockIdx.x * WAVES + wid; t < nTiles; t += gridDim.x * WAVES) {
        // Stage concat(x, agg) tile: rows of 128 (x cols 0..63, agg cols 64..127).
        const float4* xs = (const float4*)(x   + (size_t)t * 16 * D_NODE);
        const float4* gs = (const float4*)(agg + (size_t)t * 16 * D_EDGE);
#pragma unroll
        for (int i = lane; i < 256; i += 32) {
            int r = i >> 4, c = i & 15;
            *(float4*)(tA + r * 128 + c * 4)      = xs[r * 16 + c];
            *(float4*)(tA + r * 128 + 64 + c * 4) = gs[r * 16 + c];
        }

        layer128<8, true >(tA, sW0, sB0, tB, lane);   // 16x128 -> 16x128, ReLU
        layer128<8, true >(tB, sW1, sB1, tA, lane);   // 16x128 -> 16x128, ReLU
        layer128<4, false>(tA, sW2, sB2, tB, lane);   // 16x128 -> 16x64

        // LayerNorm + residual; tB holds H2 as 16x64 row-major.
        const float* xr = x   + (size_t)t * 16 * D_NODE;
        float* od       = out + (size_t)t * 16 * D_NODE;
#pragma unroll 4
        for (int r = 0; r < 16; ++r) {
            float v0 = tB[r * 64 + lane];
            float v1 = tB[r * 64 + 32 + lane];
            float s  = v0 + v1;
            float ss = v0 * v0 + v1 * v1;
#pragma unroll
            for (int o = 16; o > 0; o >>= 1) {
                s  += __shfl_xor(s,  o, 32);
                ss += __shfl_xor(ss, o, 32);
            }
            float mu  = s * (1.0f / 64.0f);
            float var = ss * (1.0f / 64.0f) - mu * mu;
            float inv = rsqrtf(fmaxf(var, 0.0f) + LN_EPS);
            od[r * 64 + lane]      = (v0 - mu) * inv * sG[lane]      + sBt[lane]      + xr[r * 64 + lane];
            od[r * 64 + 32 + lane] = (v1 - mu) * inv * sG[32 + lane] + sBt[32 + lane] + xr[r * 64 + 32 + lane];
        }
    }
}

// ---------------------------------------------------------------- launch
extern "C" void kernel_launch(void* const* d_in, const int* in_sizes, int n_in,
                              void* d_out, int out_size, void* d_ws, size_t ws_size,
                              hipStream_t stream) {
    const float*     x     = (const float*)d_in[0];
    const long long* eidx  = (const long long*)d_in[1];   // int64, [2, N_EDGES]
    const float*     eattr = (const float*)d_in[2];
    const float*     W0    = (const float*)d_in[3];
    const float*     b0    = (const float*)d_in[4];
    const float*     W1    = (const float*)d_in[5];
    const float*     b1    = (const float*)d_in[6];
    const float*     W2    = (const float*)d_in[7];
    const float*     b2    = (const float*)d_in[8];
    const float*     lng   = (const float*)d_in[9];
    const float*     lnb   = (const float*)d_in[10];
    float* out = (float*)d_out;
    float* agg = (float*)d_ws;             // N_NODES * 64 fp32 = 25.6 MB

    const int n4 = N_NODES * D_EDGE / 4;   // 1.6M float4
    zero_kernel<<<(n4 + 255) / 256, 256, 0, stream>>>((float4*)agg, n4);

    const int scatterBlocks = (N_EDGES * 32 + 255) / 256;   // one wave per edge
    scatter_kernel<<<scatterBlocks, 256, 0, stream>>>(eattr, eidx + N_EDGES, agg);

    mlp_kernel<<<625, 32 * WAVES, 0, stream>>>(x, agg, W0, b0, W1, b1,
                                               W2, b2, lng, lnb, out);
}